// GraphSAGE_87892210745355
// MI455X (gfx1250) — compile-verified
//
#include <hip/hip_runtime.h>
#include <hip/hip_bf16.h>

typedef __attribute__((ext_vector_type(2))) float v2f;
typedef __attribute__((ext_vector_type(8))) float v8f;
typedef __attribute__((ext_vector_type(4))) unsigned int u32x4;
typedef __attribute__((ext_vector_type(4))) int i32x4;
typedef __attribute__((ext_vector_type(8))) int i32x8;

#define D_HID 64

__device__ __forceinline__ v8f wmma_f32_k4(v2f a, v2f b, v8f c) {
  // V_WMMA_F32_16X16X4_F32 : D = A(16x4) * B(4x16) + C(16x16), f32
  return __builtin_amdgcn_wmma_f32_16x16x4_f32(
      /*neg_a=*/false, a, /*neg_b=*/false, b,
      /*c_mod=*/(short)0, c, /*reuse_a=*/false, /*reuse_b=*/false);
}

// ---------------------------------------------------------------- TDM load
// 2D tile load Global -> LDS via Tensor Data Mover (D# per ISA 08 ss8.3/8.4).
// dim0/tile_dim0 = row width (elems), tile1 = tile rows, tensor1 = rows
// available from gptr (OOB rows read as zero), stride0 = row stride (elems).
// pad: insert 2 dwords of LDS padding after every 64 dwords (row stride 66).
__device__ __forceinline__ void tdm_load_2d(const void* gptr, unsigned lds_byte_addr,
                                            unsigned dim0, unsigned tile1,
                                            unsigned tensor1, unsigned stride0,
                                            bool pad) {
  unsigned long long ga = (unsigned long long)(size_t)gptr;
  u32x4 g0;
  g0.x = 1u;                                              // count=1, user desc
  g0.y = lds_byte_addr;                                   // lds_addr
  g0.z = (unsigned)(ga & 0xFFFFFFFFu);                    // global_addr[31:0]
  g0.w = (unsigned)((ga >> 32) & 0x1FFFFFFu) | (2u << 30);// addr[56:32] | type=2
  i32x8 g1;
  unsigned d0 = (2u << 16);                               // data_size = 4B
  if (pad) d0 |= (1u << 20) | (5u << 22) | (1u << 25);    // pad: every 64 dw, +2 dw
  g1.s0 = (int)d0;
  g1.s1 = (int)(dim0 << 16);                              // tensor_dim0[15:0]
  g1.s2 = (int)((dim0 >> 16) | (tensor1 << 16));          // dim0[31:16] | dim1[15:0]
  g1.s3 = (int)((tensor1 >> 16) | (dim0 << 16));          // dim1[31:16] | tile_dim0
  g1.s4 = (int)(tile1 & 0xFFFFu);                         // tile_dim1 (tile_dim2=0)
  g1.s5 = (int)stride0;                                   // tensor_dim0_stride[31:0]
  g1.s6 = 0;
  g1.s7 = 0;
  i32x4 z4 = {0, 0, 0, 0};
#if defined(__clang_major__) && (__clang_major__ >= 23)
  i32x8 z8 = {0, 0, 0, 0, 0, 0, 0, 0};
  __builtin_amdgcn_tensor_load_to_lds(g0, g1, z4, z4, z8, 0);
#else
  __builtin_amdgcn_tensor_load_to_lds(g0, g1, z4, z4, 0);
#endif
}

// ---------------------------------------------------------------- utilities
__global__ void zero_kernel(float* __restrict__ p, long long n) {
  long long i = (long long)blockIdx.x * blockDim.x + threadIdx.x;
  long long stride = (long long)gridDim.x * blockDim.x;
  for (; i < n; i += stride) p[i] = 0.0f;
}

__global__ void degree_kernel(const int* __restrict__ ei, float* __restrict__ deg, int E) {
  int e = blockIdx.x * blockDim.x + threadIdx.x;
  if (e < E) atomicAdd(&deg[ei[E + e]], 1.0f);
}

// 16 threads per edge; thread c handles floats [4c,4c+4) of the 64-float row.
__global__ void scatter_kernel(const int* __restrict__ ei, const float* __restrict__ H,
                               float* __restrict__ agg, int E) {
  long long t = (long long)blockIdx.x * blockDim.x + threadIdx.x;
  if (t >= (long long)E * 16) return;
  int e = (int)(t >> 4);
  int c = (int)(t & 15);
  int s = ei[e];
  int d = ei[E + e];
  float4 v = ((const float4*)(H + (size_t)s * D_HID))[c];
  float* dst = agg + (size_t)d * D_HID + c * 4;
  atomicAdd(dst + 0, v.x);
  atomicAdd(dst + 1, v.y);
  atomicAdd(dst + 2, v.z);
  atomicAdd(dst + 3, v.w);
}

// ------------------------------------------------ SAGE layer: WMMA f32 GEMM
// Out = relu(X @ Wself + (AggSum/denom) @ Wneigh + b)
// Block = 256 threads (8 waves). Wave (mt,nt) computes a 16x16 tile:
//   rows [blockIdx.x*32 + mt*16, +16), cols [nt*16, +16).
// Operands staged in LDS by the Tensor Data Mover (wave 0), then fed to
// v_wmma_f32_16x16x4_f32 from ds_load.
__global__ void sage_layer_kernel(const float* __restrict__ X,
                                  const float* __restrict__ AggSum,
                                  const float* __restrict__ deg,
                                  const float* __restrict__ Wself,
                                  const float* __restrict__ Wneigh,
                                  const float* __restrict__ bias,
                                  float* __restrict__ Out,
                                  int nNodes) {
  // LDS: Wself[64*64] | Wneigh[64*64] | X[32 rows, stride 66] | Agg[32, 66]
  __shared__ float lds[12416];
  constexpr int W0 = 0, W1 = 4096, XO = 8192, AO = 10304;

  const int lane = threadIdx.x & 31;
  const int wave = threadIdx.x >> 5;
  const int mblk = blockIdx.x * 32;

  if (wave == 0) {
    const unsigned ldsbase = (unsigned)(size_t)(void*)lds;
    tdm_load_2d(Wself,  ldsbase + W0 * 4, 64, 64, 64, 64, false);
    tdm_load_2d(Wneigh, ldsbase + W1 * 4, 64, 64, 64, 64, false);
    const unsigned rowsRem = (unsigned)(nNodes - mblk);
    tdm_load_2d(X + (size_t)mblk * D_HID,      ldsbase + XO * 4, 64, 32, rowsRem, 64, true);
    tdm_load_2d(AggSum + (size_t)mblk * D_HID, ldsbase + AO * 4, 64, 32, rowsRem, 64, true);
    __builtin_amdgcn_s_wait_tensorcnt(0);
  }
  __syncthreads();

  const int mt = wave >> 2;           // 0..1
  const int nt = wave & 3;            // 0..3
  const int m0l = mt * 16;            // local row base
  const int n0 = nt * 16;
  const int half = lane >> 4;         // lanes 16-31 hold K+2..K+3
  const int r = lane & 15;
  const int kOff = half * 2;

  int rowG = mblk + m0l + r;
  if (rowG >= nNodes) rowG = nNodes - 1;   // clamp; stores guarded
  const float invd = 1.0f / fmaxf(deg[rowG], 1.0f);

  const float* xr = &lds[XO + (m0l + r) * 66];
  const float* ar = &lds[AO + (m0l + r) * 66];
  const float* ws = &lds[W0 + n0 + r];
  const float* wn = &lds[W1 + n0 + r];

  v8f acc = {};
#pragma unroll
  for (int k = 0; k < D_HID; k += 4) {
    v2f a, b;
    a.x = xr[k + kOff];
    a.y = xr[k + kOff + 1];
    b.x = ws[(k + kOff) * D_HID];
    b.y = ws[(k + kOff + 1) * D_HID];
    acc = wmma_f32_k4(a, b, acc);
  }
#pragma unroll
  for (int k = 0; k < D_HID; k += 4) {
    v2f a, b;
    a.x = ar[k + kOff] * invd;
    a.y = ar[k + kOff + 1] * invd;
    b.x = wn[(k + kOff) * D_HID];
    b.y = wn[(k + kOff + 1) * D_HID];
    acc = wmma_f32_k4(a, b, acc);
  }

  const float bv = bias[n0 + r];
  // C/D layout: VGPR i -> M = i + 8*half, N = lane%16
#pragma unroll
  for (int i = 0; i < 8; ++i) {
    const int row = mblk + m0l + half * 8 + i;
    if (row < nNodes)
      Out[(size_t)row * D_HID + n0 + r] = fmaxf(acc[i] + bv, 0.0f);
  }
}

// --------------------------------------------------- prototype head helpers
__global__ void hn_kernel(const float* __restrict__ H, const float* __restrict__ G,
                          float* __restrict__ hn, int n) {
  int i = blockIdx.x * blockDim.x + threadIdx.x;
  if (i >= n) return;
  const float4* h4 = (const float4*)(H + (size_t)i * D_HID);
  const float4* g4 = (const float4*)(G + (size_t)i * D_HID);
  float s = 0.0f;
#pragma unroll
  for (int j = 0; j < 16; ++j) {
    float4 v = h4[j];
    s += v.x * v.x + v.y * v.y + v.z * v.z + v.w * v.w;
  }
#pragma unroll
  for (int j = 0; j < 16; ++j) {
    float4 v = g4[j];
    s += v.x * v.x + v.y * v.y + v.z * v.z + v.w * v.w;
  }
  hn[i] = s;
}

__global__ void pn_kernel(const float* __restrict__ proto, float* __restrict__ pn, int C) {
  int c = threadIdx.x;
  if (c >= C) return;
  const float* p = proto + (size_t)c * (2 * D_HID);
  float s = 0.0f;
#pragma unroll
  for (int k = 0; k < 2 * D_HID; ++k) s += p[k] * p[k];
  pn[c] = s;
}

// logits[i][c] = 2 * <[h2,gfeat]_i, proto_c> - hn[i] - pn[c]
// Block = 96 threads (3 waves); wave w handles class tile [w*16, w*16+16).
__global__ void logits_kernel(const float* __restrict__ H,
                              const float* __restrict__ G,
                              const float* __restrict__ proto,
                              const float* __restrict__ hn,
                              const float* __restrict__ pn,
                              float* __restrict__ Out,
                              int nNodes, int nClasses) {
  const int lane = threadIdx.x & 31;
  const int wave = threadIdx.x >> 5;  // 0..2
  const int m0 = blockIdx.x * 16;
  const int n0 = wave * 16;
  const int half = lane >> 4;
  const int r = lane & 15;
  const int kOff = half * 2;

  int rowA = m0 + r;
  if (rowA >= nNodes) rowA = nNodes - 1;
  int colB = n0 + r;
  if (colB >= nClasses) colB = nClasses - 1;  // clamp loads; stores guarded

  const float* __restrict__ hrow = H + (size_t)rowA * D_HID;
  const float* __restrict__ grow = G + (size_t)rowA * D_HID;
  const float* __restrict__ prow = proto + (size_t)colB * (2 * D_HID);

  v8f acc = {};
#pragma unroll
  for (int k = 0; k < D_HID; k += 4) {
    v2f a, b;
    a.x = hrow[k + kOff];
    a.y = hrow[k + kOff + 1];
    b.x = prow[k + kOff];          // B[k][n] = proto[n][k]
    b.y = prow[k + kOff + 1];
    acc = wmma_f32_k4(a, b, acc);
  }
#pragma unroll
  for (int k = 0; k < D_HID; k += 4) {
    v2f a, b;
    a.x = grow[k + kOff];
    a.y = grow[k + kOff + 1];
    b.x = prow[D_HID + k + kOff];
    b.y = prow[D_HID + k + kOff + 1];
    acc = wmma_f32_k4(a, b, acc);
  }

  const int col = n0 + r;
  if (col < nClasses) {
    const float pnv = pn[col];
#pragma unroll
    for (int i = 0; i < 8; ++i) {
      const int row = m0 + half * 8 + i;
      if (row < nNodes)
        Out[(size_t)row * nClasses + col] = 2.0f * acc[i] - hn[row] - pnv;
    }
  }
}

// ----------------------------------------------------------------- launcher
extern "C" void kernel_launch(void* const* d_in, const int* in_sizes, int n_in,
                              void* d_out, int out_size, void* d_ws, size_t ws_size,
                              hipStream_t stream) {
  const float* feat  = (const float*)d_in[0];
  const float* gfeat = (const float*)d_in[1];
  const int*   ei    = (const int*)d_in[2];
  const float* Ws0   = (const float*)d_in[3];
  const float* Wn0   = (const float*)d_in[4];
  const float* b0    = (const float*)d_in[5];
  const float* Ws1   = (const float*)d_in[6];
  const float* Wn1   = (const float*)d_in[7];
  const float* b1    = (const float*)d_in[8];
  const float* proto = (const float*)d_in[9];
  float* out = (float*)d_out;

  const int N = in_sizes[0] / D_HID;         // 100000
  const int E = in_sizes[2] / 2;             // 1600000
  const int C = in_sizes[9] / (2 * D_HID);   // 40

  float* ws  = (float*)d_ws;
  float* deg = ws;                                // N
  float* agg = deg + N;                           // N*64
  float* h1  = agg + (size_t)N * D_HID;           // N*64
  float* h2  = h1 + (size_t)N * D_HID;            // N*64
  float* hn  = h2 + (size_t)N * D_HID;            // N
  float* pn  = hn + N;                            // C (pad 64)

  const int scatterBlocks = (int)(((long long)E * 16 + 255) / 256);

  // deg + agg are contiguous: zero both in one pass
  zero_kernel<<<2048, 256, 0, stream>>>(deg, (long long)N * (D_HID + 1));
  degree_kernel<<<(E + 255) / 256, 256, 0, stream>>>(ei, deg, E);

  // layer 0
  scatter_kernel<<<scatterBlocks, 256, 0, stream>>>(ei, feat, agg, E);
  sage_layer_kernel<<<(N + 31) / 32, 256, 0, stream>>>(feat, agg, deg, Ws0, Wn0, b0, h1, N);

  // layer 1 (reuse agg)
  zero_kernel<<<2048, 256, 0, stream>>>(agg, (long long)N * D_HID);
  scatter_kernel<<<scatterBlocks, 256, 0, stream>>>(ei, h1, agg, E);
  sage_layer_kernel<<<(N + 31) / 32, 256, 0, stream>>>(h1, agg, deg, Ws1, Wn1, b1, h2, N);

  // prototype head
  hn_kernel<<<(N + 255) / 256, 256, 0, stream>>>(h2, gfeat, hn, N);
  pn_kernel<<<1, 64, 0, stream>>>(proto, pn, C);
  logits_kernel<<<(N + 15) / 16, 96, 0, stream>>>(h2, gfeat, proto, hn, pn, out, N, C);
}